// ClustUResNetEdgeEncoder_57947698758353
// MI455X (gfx1250) — compile-verified
//
#include <hip/hip_runtime.h>
#include <hip/hip_bf16.h>

typedef __attribute__((ext_vector_type(2))) float v2f;
typedef __attribute__((ext_vector_type(8))) float v8f;

// One 16-point x 16-feature tile per wave iteration.
// out[p][n] = relu( sum_k x[p][k] * W[k][n] ),  x = (d0,d1,d2,edge_id,d4)
//
// V_WMMA_F32_16X16X4_F32 layouts (wave32, per ISA 7.12.2):
//   A (16x4):  lane 0-15: M=lane,    v0->K=0, v1->K=1
//              lane 16-31: M=lane-16, v0->K=2, v1->K=3
//   B (4x16):  v, laneHalf -> K = 2*laneHalf + v, N = lane%16 (row-striped)
//   C/D (16x16): VGPR i, lanes 0-15 -> M=i, lanes 16-31 -> M=i+8, N = lane%16
__global__ __launch_bounds__(256) void edge_encoder_wmma(
    const float* __restrict__ data,
    const int*   __restrict__ clusts,
    const int*   __restrict__ edge_index,
    const float* __restrict__ W,
    float*       __restrict__ out,
    int nEdges, int clustSize, int numTiles)
{
    const int lane   = threadIdx.x & 31;
    const int laneHi = lane >> 4;   // 0: lanes 0-15, 1: lanes 16-31
    const int laneLo = lane & 15;
    const int waveId   = (blockIdx.x * blockDim.x + threadIdx.x) >> 5;
    const int numWaves = (gridDim.x * blockDim.x) >> 5;

    // B1: K-slots 0..3 = W rows 0..3 (feature 3 is the edge-id column).
    // B2: K-slot 0 = W row 4; K-slots 1..3 zero (K padding).
    v2f b1, b2;
    b1.x = W[(2 * laneHi + 0) * 16 + laneLo];
    b1.y = W[(2 * laneHi + 1) * 16 + laneLo];
    b2.x = laneHi ? 0.0f : W[4 * 16 + laneLo];
    b2.y = 0.0f;

    const int rowsPerEdge = 2 * clustSize;
    const bool hi = (laneHi != 0);
    const float inf = __builtin_inff();

    for (int tile = waveId; tile < numTiles; tile += numWaves) {
        // Both lane-halves address the same point M = laneLo of this tile.
        int m    = tile * 16 + laneLo;
        int e    = m / rowsPerEdge;
        int r    = m - e * rowsPerEdge;
        int side = (r >= clustSize) ? 1 : 0;
        int v    = r - side * clustSize;
        int c    = edge_index[side * nEdges + e];   // [2, E] row-major
        int vox  = clusts[c * clustSize + v];
        const float* row = data + (long long)vox * 5;

        // Unconditional loads (data is L2-resident), lane-half selects.
        float d0 = row[0];
        float d1 = row[1];
        float d2 = row[2];
        float d4 = row[4];

        v2f a1, a2;
        a1.x = hi ? d2 : d0;          // K=0: feat0   | K=2: feat2
        a1.y = hi ? (float)e : d1;    // K=1: feat1   | K=3: edge-id
        a2.x = hi ? 0.0f : d4;        // K=0: feat4   | K=2: pad
        a2.y = 0.0f;                  // K=1 / K=3 pad

        v8f acc = {};
        acc = __builtin_amdgcn_wmma_f32_16x16x4_f32(
            false, a1, false, b1, (short)0, acc, false, false);
        acc = __builtin_amdgcn_wmma_f32_16x16x4_f32(
            false, a2, false, b2, (short)0, acc, false, false);

        // Store with ReLU (v_med3_f32 clamp -> single VALU op per element).
        // For each VGPR i, lanes 0-15 and 16-31 each cover a contiguous 64B
        // feature row -> coalesced. Output (410 MB) streams past L2:
        // non-temporal stores. All 16 rows of a processed tile are in-range
        // by construction, so no per-element guard.
        float* dst = out + ((long long)(tile * 16 + 8 * laneHi)) * 16 + laneLo;
        #pragma unroll
        for (int i = 0; i < 8; ++i) {
            float val = __builtin_amdgcn_fmed3f(acc[i], 0.0f, inf);
            __builtin_nontemporal_store(val, dst + i * 16);
        }
    }
}

// Scalar tail for numPoints not divisible by 16 (not hit for the reference
// shapes, where numPoints = E*2*V = 6.4M is a multiple of 16).
__global__ void edge_encoder_tail(
    const float* __restrict__ data,
    const int*   __restrict__ clusts,
    const int*   __restrict__ edge_index,
    const float* __restrict__ W,
    float*       __restrict__ out,
    int nEdges, int clustSize, int firstPoint, int numPoints)
{
    int p = firstPoint + (blockIdx.x * blockDim.x + threadIdx.x) / 16;
    int n = threadIdx.x & 15;
    if (p >= numPoints) return;
    int rowsPerEdge = 2 * clustSize;
    int e    = p / rowsPerEdge;
    int r    = p - e * rowsPerEdge;
    int side = (r >= clustSize) ? 1 : 0;
    int v    = r - side * clustSize;
    int c    = edge_index[side * nEdges + e];
    int vox  = clusts[c * clustSize + v];
    const float* row = data + (long long)vox * 5;
    float acc = row[0] * W[0 * 16 + n] + row[1] * W[1 * 16 + n] +
                row[2] * W[2 * 16 + n] + (float)e * W[3 * 16 + n] +
                row[4] * W[4 * 16 + n];
    out[(long long)p * 16 + n] = fmaxf(acc, 0.0f);
}

extern "C" void kernel_launch(void* const* d_in, const int* in_sizes, int n_in,
                              void* d_out, int out_size, void* d_ws, size_t ws_size,
                              hipStream_t stream) {
    const float* data       = (const float*)d_in[0];
    const int*   clusts     = (const int*)d_in[1];
    const int*   edge_index = (const int*)d_in[2];
    const float* W          = (const float*)d_in[3];
    float*       out        = (float*)d_out;

    const int nEdges    = in_sizes[2] / 2;                 // edge_index is [2, E]
    const int numPoints = out_size / 16;                   // F = 16 features
    const int clustSize = numPoints / (2 * nEdges);        // = 100
    const int numTiles  = numPoints / 16;

    dim3 block(256);
    dim3 grid(4096);
    edge_encoder_wmma<<<grid, block, 0, stream>>>(
        data, clusts, edge_index, W, out, nEdges, clustSize, numTiles);

    const int tailStart = numTiles * 16;
    const int tailPts   = numPoints - tailStart;
    if (tailPts > 0) {
        int threads = tailPts * 16;
        edge_encoder_tail<<<(threads + 255) / 256, 256, 0, stream>>>(
            data, clusts, edge_index, W, out, nEdges, clustSize,
            tailStart, numPoints);
    }
}